// MultiQueryAttention_32263794327903
// MI455X (gfx1250) — compile-verified
//
#include <hip/hip_runtime.h>
#include <math.h>

typedef __attribute__((ext_vector_type(2))) float v2f;
typedef __attribute__((ext_vector_type(8))) float v8f;

#define WMMA_F32(a, b, c) \
    __builtin_amdgcn_wmma_f32_16x16x4_f32(false, (a), false, (b), (short)0, (c), false, false)

// ---------------------------------------------------------------------------
// Generic fp32 WMMA GEMM:  C[M,N] = alpha * (A[M,K] @ W[K,N]) + bias[N]
// Block tile 128x64, 256 threads = 8 waves arranged 4(m) x 2(n);
// each wave owns a 32x32 tile (4 x 16x16 f32 accumulators).
// W tile stored transposed in LDS so every A/B fragment is one ds_load_b64.
// Row pads are even (18 floats) -> 8B-aligned pairs, conflict-free banks.
// ---------------------------------------------------------------------------
__global__ __launch_bounds__(256) void gemm_wmma_f32(
    const float* __restrict__ A, const float* __restrict__ W,
    const float* __restrict__ bias, float* __restrict__ C,
    int M, int N, int K, float alpha)
{
    __shared__ __align__(16) float As[128][18];  // As[m][k]
    __shared__ __align__(16) float Wt[64][18];   // Wt[n][k]  (transposed W tile)

    const int tid    = threadIdx.x;
    const int wave   = tid >> 5;
    const int lane   = tid & 31;
    const int l16    = lane & 15;
    const int half   = lane >> 4;
    const int kb     = half << 1;   // fragment K offset (0 or 2) -> even
    const int rowOff = half << 3;   // C row offset (0 or 8)

    const int wm = wave >> 1;       // 0..3 : 32-row group
    const int wn = wave & 1;        // 0..1 : 32-col group

    const int rowBlock = blockIdx.y * 128;
    const int colBlock = blockIdx.x * 64;

    v8f acc[2][2];
    #pragma unroll
    for (int mt = 0; mt < 2; ++mt)
        #pragma unroll
        for (int nt = 0; nt < 2; ++nt) acc[mt][nt] = (v8f)(0.0f);

    const int ar = tid >> 1;          // 0..127 : A row in tile
    const int ac = (tid & 1) << 3;    // 0 or 8 : A col half
    const int wr = tid >> 4;          // 0..15  : W k-row in tile
    const int wc = (tid & 15) << 2;   // 0..60  : W col group

    const int aRow0 = (wm << 5) + l16;
    const int aRow1 = aRow0 + 16;
    const int bRow0 = (wn << 5) + l16;
    const int bRow1 = bRow0 + 16;

    for (int kt = 0; kt < K; kt += 16) {
        // load A tile 128x16 (2x float4 per thread), row-major
        const float* aSrc = A + (size_t)(rowBlock + ar) * K + kt + ac;
        const float4 a0 = *(const float4*)(aSrc);
        const float4 a1 = *(const float4*)(aSrc + 4);
        As[ar][ac + 0] = a0.x; As[ar][ac + 1] = a0.y;
        As[ar][ac + 2] = a0.z; As[ar][ac + 3] = a0.w;
        As[ar][ac + 4] = a1.x; As[ar][ac + 5] = a1.y;
        As[ar][ac + 6] = a1.z; As[ar][ac + 7] = a1.w;
        // load W tile 16x64 (1x float4 per thread), store transposed
        const float* wSrc = W + (size_t)(kt + wr) * N + colBlock + wc;
        const float4 wv = *(const float4*)(wSrc);
        Wt[wc + 0][wr] = wv.x; Wt[wc + 1][wr] = wv.y;
        Wt[wc + 2][wr] = wv.z; Wt[wc + 3][wr] = wv.w;
        if (kt + 16 < K) {                        // gfx1250 global_prefetch_b8
            __builtin_prefetch(aSrc + 16, 0, 0);
            __builtin_prefetch(wSrc + (size_t)16 * N, 0, 0);
        }
        __syncthreads();

        #pragma unroll
        for (int kk = 0; kk < 16; kk += 4) {
            const v2f av0 = *(const v2f*)&As[aRow0][kk + kb];
            const v2f av1 = *(const v2f*)&As[aRow1][kk + kb];
            const v2f bv0 = *(const v2f*)&Wt[bRow0][kk + kb];
            const v2f bv1 = *(const v2f*)&Wt[bRow1][kk + kb];
            acc[0][0] = WMMA_F32(av0, bv0, acc[0][0]);
            acc[0][1] = WMMA_F32(av0, bv1, acc[0][1]);
            acc[1][0] = WMMA_F32(av1, bv0, acc[1][0]);
            acc[1][1] = WMMA_F32(av1, bv1, acc[1][1]);
        }
        __syncthreads();
    }

    #pragma unroll
    for (int mt = 0; mt < 2; ++mt) {
        #pragma unroll
        for (int nt = 0; nt < 2; ++nt) {
            const int col = colBlock + (wn << 5) + (nt << 4) + l16;
            const float bv = bias ? bias[col] : 0.0f;
            #pragma unroll
            for (int v = 0; v < 8; ++v) {
                const int row = rowBlock + (wm << 5) + (mt << 4) + rowOff + v;
                C[(size_t)row * N + col] = alpha * acc[mt][nt][v] + bv;
            }
        }
    }
}

// ---------------------------------------------------------------------------
// Flash attention (causal, MQA: single shared KV head of 64).
// Grid: (S/64, NUM_HEADS, BATCH). Block: 128 threads = 4 waves.
// Each wave owns 16 query rows x 64 output dims. Q fragments kept in VGPRs.
// V tile stored transposed so P@V B-fragments are single ds_load_b64.
// Q pre-scaled by 1/sqrt(64) in the projection GEMM.
// ---------------------------------------------------------------------------
__global__ __launch_bounds__(128) void mqa_flash_wmma(
    const float* __restrict__ Q,    // [B*S, DIM], head h at cols h*64
    const float* __restrict__ Kg,   // [B*S, 64]
    const float* __restrict__ Vg,   // [B*S, 64]
    float* __restrict__ Out,        // [B*S, DIM]
    int S, int DIMp)
{
    __shared__ __align__(16) float Ks[64][68];   // Ks[key][d]
    __shared__ __align__(16) float Vt[64][68];   // Vt[d][key] (transposed)
    __shared__ __align__(16) float Ps[64][68];   // Q staging, then P

    const int tid    = threadIdx.x;
    const int wave   = tid >> 5;
    const int lane   = tid & 31;
    const int l16    = lane & 15;
    const int half   = lane >> 4;
    const int kb     = half << 1;
    const int rowOff = half << 3;

    const int qTile = blockIdx.x;
    const int h     = blockIdx.y;
    const int b     = blockIdx.z;
    const int qBase = qTile << 6;
    const size_t rowB = (size_t)b * S;

    const int lr = tid >> 1;          // 0..63 tile row for cooperative loads
    const int lc = (tid & 1) << 5;    // 0 or 32

    // --- stage Q tile into Ps, then pull A-fragments into registers ---
    {
        const float* src = Q + (rowB + qBase + lr) * DIMp + (h << 6) + lc;
        #pragma unroll
        for (int j = 0; j < 8; ++j) {
            const float4 qv = *(const float4*)(src + (j << 2));
            Ps[lr][lc + (j << 2) + 0] = qv.x; Ps[lr][lc + (j << 2) + 1] = qv.y;
            Ps[lr][lc + (j << 2) + 2] = qv.z; Ps[lr][lc + (j << 2) + 3] = qv.w;
        }
    }
    __syncthreads();

    v2f aq[16];
    {
        const int qr = (wave << 4) + l16;
        #pragma unroll
        for (int i = 0; i < 16; ++i)
            aq[i] = *(const v2f*)&Ps[qr][(i << 2) + kb];
    }

    v8f o[4];
    #pragma unroll
    for (int nt = 0; nt < 4; ++nt) o[nt] = (v8f)(0.0f);
    float m[8], lsum[8];
    #pragma unroll
    for (int v = 0; v < 8; ++v) { m[v] = -1e30f; lsum[v] = 0.0f; }

    const int nTiles = qTile + 1;     // causal: key tiles 0..qTile
    for (int t = 0; t < nTiles; ++t) {
        const int kBase = t << 6;
        __syncthreads();              // protect Ks/Vt/Ps reuse across iterations
        {
            const float* ksrc = Kg + (rowB + kBase + lr) * 64 + lc;
            const float* vsrc = Vg + (rowB + kBase + lr) * 64 + lc;
            #pragma unroll
            for (int j = 0; j < 8; ++j) {
                const float4 kv = *(const float4*)(ksrc + (j << 2));
                const float4 vv = *(const float4*)(vsrc + (j << 2));
                Ks[lr][lc + (j << 2) + 0] = kv.x; Ks[lr][lc + (j << 2) + 1] = kv.y;
                Ks[lr][lc + (j << 2) + 2] = kv.z; Ks[lr][lc + (j << 2) + 3] = kv.w;
                Vt[lc + (j << 2) + 0][lr] = vv.x; Vt[lc + (j << 2) + 1][lr] = vv.y;
                Vt[lc + (j << 2) + 2][lr] = vv.z; Vt[lc + (j << 2) + 3][lr] = vv.w;
            }
        }
        __syncthreads();

        // ---- S = Q @ K^T (16 x 64 per wave) ----
        v8f sc[4];
        #pragma unroll
        for (int nt = 0; nt < 4; ++nt) sc[nt] = (v8f)(0.0f);
        #pragma unroll
        for (int kk = 0; kk < 64; kk += 4) {
            const v2f a = aq[kk >> 2];
            #pragma unroll
            for (int nt = 0; nt < 4; ++nt) {
                const v2f bf = *(const v2f*)&Ks[(nt << 4) + l16][kk + kb];
                sc[nt] = WMMA_F32(a, bf, sc[nt]);
            }
        }

        // ---- causal mask (diagonal tile only) ----
        if (t == nTiles - 1) {
            #pragma unroll
            for (int nt = 0; nt < 4; ++nt) {
                const int key = kBase + (nt << 4) + l16;
                #pragma unroll
                for (int v = 0; v < 8; ++v) {
                    const int qr = qBase + (wave << 4) + rowOff + v;
                    if (key > qr) sc[nt][v] = -1e30f;
                }
            }
        }

        // ---- online softmax (row reductions within 16-lane C-layout groups) ----
        float corr[8];
        #pragma unroll
        for (int v = 0; v < 8; ++v) {
            float x = fmaxf(fmaxf(sc[0][v], sc[1][v]), fmaxf(sc[2][v], sc[3][v]));
            x = fmaxf(x, __shfl_xor(x, 1, 32));
            x = fmaxf(x, __shfl_xor(x, 2, 32));
            x = fmaxf(x, __shfl_xor(x, 4, 32));
            x = fmaxf(x, __shfl_xor(x, 8, 32));
            const float mNew = fmaxf(m[v], x);
            corr[v] = __expf(m[v] - mNew);
            m[v] = mNew;
        }
        #pragma unroll
        for (int nt = 0; nt < 4; ++nt)
            #pragma unroll
            for (int v = 0; v < 8; ++v) o[nt][v] *= corr[v];

        float rs[8];
        #pragma unroll
        for (int v = 0; v < 8; ++v) rs[v] = 0.0f;
        const int pRowBase = (wave << 4) + rowOff;
        #pragma unroll
        for (int nt = 0; nt < 4; ++nt) {
            #pragma unroll
            for (int v = 0; v < 8; ++v) {
                const float p = __expf(sc[nt][v] - m[v]);
                rs[v] += p;
                Ps[pRowBase + v][(nt << 4) + l16] = p;  // C-layout -> A-layout via LDS
            }
        }
        #pragma unroll
        for (int v = 0; v < 8; ++v) {
            float r = rs[v];
            r += __shfl_xor(r, 1, 32);
            r += __shfl_xor(r, 2, 32);
            r += __shfl_xor(r, 4, 32);
            r += __shfl_xor(r, 8, 32);
            lsum[v] = lsum[v] * corr[v] + r;
        }
        __syncthreads();

        // ---- O += P @ V ----
        const int pr = (wave << 4) + l16;
        #pragma unroll
        for (int kk = 0; kk < 64; kk += 4) {
            const v2f a = *(const v2f*)&Ps[pr][kk + kb];
            #pragma unroll
            for (int nt = 0; nt < 4; ++nt) {
                const v2f bf = *(const v2f*)&Vt[(nt << 4) + l16][kk + kb];
                o[nt] = WMMA_F32(a, bf, o[nt]);
            }
        }
    }

    // ---- normalize + store in [b, s, h*64 + c] layout (no transpose needed) ----
    #pragma unroll
    for (int nt = 0; nt < 4; ++nt) {
        const int col = (h << 6) + (nt << 4) + l16;
        #pragma unroll
        for (int v = 0; v < 8; ++v) {
            const int row = qBase + (wave << 4) + rowOff + v;
            Out[(rowB + row) * DIMp + col] = o[nt][v] / lsum[v];
        }
    }
}

// ---------------------------------------------------------------------------
// Host launcher
// ---------------------------------------------------------------------------
extern "C" void kernel_launch(void* const* d_in, const int* in_sizes, int n_in,
                              void* d_out, int out_size, void* d_ws, size_t ws_size,
                              hipStream_t stream) {
    (void)in_sizes; (void)n_in; (void)out_size; (void)ws_size;
    constexpr int DIM = 1024, HEAD_DIM = 64, NUM_HEADS = 16, BATCH = 2, SEQ = 2048;
    constexpr int M = BATCH * SEQ;                       // 4096

    const float* x  = (const float*)d_in[0];
    const float* Wq = (const float*)d_in[1];
    const float* Wk = (const float*)d_in[2];
    const float* Wv = (const float*)d_in[3];
    const float* Wo = (const float*)d_in[4];
    const float* bo = (const float*)d_in[5];
    float* out = (float*)d_out;

    float* Qbuf = (float*)d_ws;                          // [4096,1024]
    float* Kbuf = Qbuf + (size_t)M * DIM;                // [4096,64]
    float* Vbuf = Kbuf + (size_t)M * HEAD_DIM;           // [4096,64]
    float* Abuf = Vbuf + (size_t)M * HEAD_DIM;           // [4096,1024]

    const float scale = 0.125f;                          // 1/sqrt(64), folded into Q

    gemm_wmma_f32<<<dim3(DIM / 64, M / 128), 256, 0, stream>>>(
        x, Wq, nullptr, Qbuf, M, DIM, DIM, scale);
    gemm_wmma_f32<<<dim3(HEAD_DIM / 64, M / 128), 256, 0, stream>>>(
        x, Wk, nullptr, Kbuf, M, HEAD_DIM, DIM, 1.0f);
    gemm_wmma_f32<<<dim3(HEAD_DIM / 64, M / 128), 256, 0, stream>>>(
        x, Wv, nullptr, Vbuf, M, HEAD_DIM, DIM, 1.0f);

    mqa_flash_wmma<<<dim3(SEQ / 64, NUM_HEADS, BATCH), 128, 0, stream>>>(
        Qbuf, Kbuf, Vbuf, Abuf, SEQ, DIM);

    gemm_wmma_f32<<<dim3(DIM / 64, M / 128), 256, 0, stream>>>(
        Abuf, Wo, bo, out, M, DIM, DIM, 1.0f);
}